// GraphEncoder_75531294867867
// MI455X (gfx1250) — compile-verified
//
#include <hip/hip_runtime.h>
#include <math.h>

#define F_IN 128
#define HDIM 256

typedef float v2f __attribute__((ext_vector_type(2)));
typedef float v8f __attribute__((ext_vector_type(8)));

// ---------------------------------------------------------------------------
// 1) Row L2-normalize: one wave32 per row of 128 floats (4 per lane).
// ---------------------------------------------------------------------------
__global__ void k_normalize(const float* __restrict__ x,
                            float* __restrict__ xn, int n) {
  int wave = (int)((blockIdx.x * blockDim.x + threadIdx.x) >> 5);
  int lane = threadIdx.x & 31;
  if (wave >= n) return;
  const float* row = x + (size_t)wave * F_IN;
  float a = row[lane], b = row[lane + 32], c = row[lane + 64], d = row[lane + 96];
  float s = a * a + b * b + c * c + d * d;
  #pragma unroll
  for (int m = 16; m >= 1; m >>= 1) s += __shfl_xor(s, m, 32);
  float scale = 1.0f / fmaxf(sqrtf(s), 1e-12f);
  float* o = xn + (size_t)wave * F_IN;
  o[lane] = a * scale; o[lane + 32] = b * scale;
  o[lane + 64] = c * scale; o[lane + 96] = d * scale;
}

// ---------------------------------------------------------------------------
// 2) Zero fill
// ---------------------------------------------------------------------------
__global__ void k_fill0(float* __restrict__ p, size_t n) {
  size_t i = (size_t)blockIdx.x * blockDim.x + threadIdx.x;
  if (i < n) p[i] = 0.0f;
}

// ---------------------------------------------------------------------------
// 3) In-degree count via float atomics (matches segment_sum of ones)
// ---------------------------------------------------------------------------
__global__ void k_degree(const int* __restrict__ dst,
                         float* __restrict__ cnt, int e) {
  int i = blockIdx.x * blockDim.x + threadIdx.x;
  if (i < e) atomicAdd(&cnt[dst[i]], 1.0f);
}

// ---------------------------------------------------------------------------
// 4) dinv = rsqrt(deg + 1) in place, both edge sets at once
// ---------------------------------------------------------------------------
__global__ void k_dinv(float* __restrict__ a, float* __restrict__ b, int n) {
  int i = blockIdx.x * blockDim.x + threadIdx.x;
  if (i < n) {
    a[i] = rsqrtf(a[i] + 1.0f);
    b[i] = rsqrtf(b[i] + 1.0f);
  }
}

// ---------------------------------------------------------------------------
// 5) xw = xn @ W1 via V_WMMA_F32_16X16X4_F32.
//    Block = 256 threads (8 waves) handles a 16-row M tile x all 256 cols.
//    Each wave owns two 16x16 N tiles; K loop of 32 steps (K=4 each).
//    A tile staged in LDS with stride 129 (conflict-free column reads).
//    Fragment layouts per CDNA5 ISA 7.12.2:
//      A (16x4 f32): lanes 0-15 hold K=0,1; lanes 16-31 hold K=2,3; M = lane&15
//      B (4x16 f32): lanes 0-15 hold K=0,1; lanes 16-31 hold K=2,3; N = lane&15
//      C/D: VGPR r -> M = r + 8*(lane>>4), N = lane&15
// ---------------------------------------------------------------------------
__global__ __launch_bounds__(256) void
k_gemm_wmma(const float* __restrict__ xn, const float* __restrict__ W,
            float* __restrict__ xw, int n) {
  __shared__ float lds[16 * 129];
  const int row0 = blockIdx.x * 16;
  const int tid = threadIdx.x;

  for (int i = tid; i < 16 * F_IN; i += 256) {
    int m = i >> 7, k = i & 127;
    int r = row0 + m;
    lds[m * 129 + k] = (r < n) ? xn[(size_t)r * F_IN + k] : 0.0f;
  }
  __syncthreads();

  const int wave  = tid >> 5;
  const int lane  = tid & 31;
  const int mlow  = lane & 15;          // M (for A) / N (for B,C)
  const int khalf = (lane >> 4) << 1;   // 0 or 2
  const int n0    = wave * 32;          // two adjacent 16-wide N tiles

  v8f c0 = {}; v8f c1 = {};
  #pragma unroll
  for (int k = 0; k < F_IN; k += 4) {
    const int kb = k + khalf;
    v2f a;
    a.x = lds[mlow * 129 + kb];
    a.y = lds[mlow * 129 + kb + 1];
    const float* wp = W + (size_t)kb * HDIM;   // W1 row-major [128][256]
    v2f b0, b1;
    b0.x = wp[n0 + mlow];            b0.y = wp[HDIM + n0 + mlow];
    b1.x = wp[n0 + 16 + mlow];       b1.y = wp[HDIM + n0 + 16 + mlow];
    c0 = __builtin_amdgcn_wmma_f32_16x16x4_f32(false, a, false, b0,
                                               (short)0, c0, false, false);
    c1 = __builtin_amdgcn_wmma_f32_16x16x4_f32(false, a, false, b1,
                                               (short)0, c1, false, false);
  }

  const int mhi = (lane >> 4) << 3;    // 0 or 8
  #pragma unroll
  for (int r = 0; r < 8; ++r) {
    int m = row0 + r + mhi;
    if (m < n) {
      xw[(size_t)m * HDIM + n0 + mlow]      = c0[r];
      xw[(size_t)m * HDIM + n0 + 16 + mlow] = c1[r];
    }
  }
}

// ---------------------------------------------------------------------------
// 6) Fused init of both outputs: out = dinv^2 * xw + b1 (self-loop + bias)
// ---------------------------------------------------------------------------
__global__ void k_selfinit(const float* __restrict__ xw,
                           const float* __restrict__ dinv_t,
                           const float* __restrict__ dinv_l,
                           const float* __restrict__ b1,
                           float* __restrict__ out_t,
                           float* __restrict__ out_l, size_t total) {
  size_t i = (size_t)blockIdx.x * blockDim.x + threadIdx.x;
  if (i >= total) return;
  size_t row = i >> 8;       // /HDIM
  int    col = (int)(i & 255);
  float v  = xw[i];
  float bb = b1[col];
  float dt = dinv_t[row];
  float dl = dinv_l[row];
  out_t[i] = dt * dt * v + bb;
  out_l[i] = dl * dl * v + bb;
}

// ---------------------------------------------------------------------------
// 7) Edge scatter: wave per edge, lane j covers cols lane+32*j (coalesced
//    128B lines for both the gather of xw[src] and the atomics into out[dst]).
//    xw (102 MB) and out slab (102 MB) are L2-resident on the 192MB L2.
// ---------------------------------------------------------------------------
__global__ void k_scatter(const float* __restrict__ xw,
                          const float* __restrict__ dinv,
                          const int* __restrict__ src,
                          const int* __restrict__ dst,
                          float* __restrict__ out, int e) {
  int gw   = (int)((blockIdx.x * blockDim.x + threadIdx.x) >> 5);
  int lane = threadIdx.x & 31;
  int nw   = (int)((gridDim.x * blockDim.x) >> 5);
  for (int eid = gw; eid < e; eid += nw) {
    int s = src[eid], d = dst[eid];
    float norm = dinv[s] * dinv[d];
    const float* xs = xw + (size_t)s * HDIM;
    float* od = out + (size_t)d * HDIM;
    #pragma unroll
    for (int j = 0; j < 8; ++j) {
      int col = lane + 32 * j;
      atomicAdd(&od[col], norm * xs[col]);
    }
  }
}

// ---------------------------------------------------------------------------
// 8) aedge = sigmoid(dot(ztop[src], ztop[dst])): wave per edge, 8 cols/lane,
//    shfl_xor tree reduction.
// ---------------------------------------------------------------------------
__global__ void k_aedge(const float* __restrict__ z,
                        const int* __restrict__ src,
                        const int* __restrict__ dst,
                        float* __restrict__ aedge, int e) {
  int gw   = (int)((blockIdx.x * blockDim.x + threadIdx.x) >> 5);
  int lane = threadIdx.x & 31;
  int nw   = (int)((gridDim.x * blockDim.x) >> 5);
  for (int eid = gw; eid < e; eid += nw) {
    int s = src[eid], d = dst[eid];
    const float* zs = z + (size_t)s * HDIM;
    const float* zd = z + (size_t)d * HDIM;
    float acc = 0.0f;
    #pragma unroll
    for (int j = 0; j < 8; ++j) {
      int col = lane + 32 * j;
      acc += zs[col] * zd[col];
    }
    #pragma unroll
    for (int m = 16; m >= 1; m >>= 1) acc += __shfl_xor(acc, m, 32);
    if (lane == 0) aedge[eid] = 1.0f / (1.0f + expf(-acc));
  }
}

// ---------------------------------------------------------------------------
extern "C" void kernel_launch(void* const* d_in, const int* in_sizes, int n_in,
                              void* d_out, int out_size, void* d_ws, size_t ws_size,
                              hipStream_t stream) {
  const float* x      = (const float*)d_in[0];
  const int*   ei_top = (const int*)d_in[1];   // [2, E]
  const int*   ei_lst = (const int*)d_in[2];   // [2, E]
  const float* W1     = (const float*)d_in[3]; // [128, 256] row-major
  const float* b1     = (const float*)d_in[4]; // [256]

  const int N = in_sizes[0] / F_IN;
  const int E = in_sizes[1] / 2;

  const int* src_t = ei_top;       const int* dst_t = ei_top + E;
  const int* src_l = ei_lst;       const int* dst_l = ei_lst + E;

  float* out_t = (float*)d_out;                       // ztop  [N, 256]
  float* out_l = out_t + (size_t)N * HDIM;            // zlast [N, 256]
  float* ae    = out_l + (size_t)N * HDIM;            // aedge [E]

  float* ws    = (float*)d_ws;
  float* xn    = ws;                                  // N*128
  float* xw    = xn + (size_t)N * F_IN;               // N*256
  float* cnt_t = xw + (size_t)N * HDIM;               // N (becomes dinv_t)
  float* cnt_l = cnt_t + N;                           // N (becomes dinv_l)

  // 1) normalize: 8 waves per 256-block
  k_normalize<<<(N + 7) / 8, 256, 0, stream>>>(x, xn, N);

  // 2) zero both degree counters (contiguous 2N floats)
  size_t twoN = 2 * (size_t)N;
  k_fill0<<<(int)((twoN + 255) / 256), 256, 0, stream>>>(cnt_t, twoN);

  // 3) degree counts
  k_degree<<<(E + 255) / 256, 256, 0, stream>>>(dst_t, cnt_t, E);
  k_degree<<<(E + 255) / 256, 256, 0, stream>>>(dst_l, cnt_l, E);

  // 4) dinv = rsqrt(deg + 1), in place
  k_dinv<<<(N + 255) / 256, 256, 0, stream>>>(cnt_t, cnt_l, N);

  // 5) WMMA GEMM xw = xn @ W1
  k_gemm_wmma<<<(N + 15) / 16, 256, 0, stream>>>(xn, W1, xw, N);

  // 6) out = dinv^2 * xw + b1 for both outputs
  size_t total = (size_t)N * HDIM;
  k_selfinit<<<(int)((total + 255) / 256), 256, 0, stream>>>(
      xw, cnt_t, cnt_l, b1, out_t, out_l, total);

  // 7) edge scatter into both outputs (wave per edge)
  k_scatter<<<(E + 7) / 8, 256, 0, stream>>>(xw, cnt_t, src_t, dst_t, out_t, E);
  k_scatter<<<(E + 7) / 8, 256, 0, stream>>>(xw, cnt_l, src_l, dst_l, out_l, E);

  // 8) aedge from completed ztop
  k_aedge<<<(E + 7) / 8, 256, 0, stream>>>(out_t, src_t, dst_t, ae, E);
}